// Customize_GCN_76871324664156
// MI455X (gfx1250) — compile-verified
//
#include <hip/hip_runtime.h>

#define N_NODES   100000
#define N_EDGES   1600000
#define IN_F      128
#define HID       64
#define N_CLASSES 40

typedef float v2f __attribute__((ext_vector_type(2)));
typedef float v8f __attribute__((ext_vector_type(8)));

// ---------------------------------------------------------------------------
// Float atomic max via sign-aware integer atomics (lowers to native
// global_atomic_max_i32 / global_atomic_min_u32; no CAS loop).
// Valid because agg is pre-initialized with finite self-loop messages.
// ---------------------------------------------------------------------------
__device__ __forceinline__ void atomicMaxF(float* addr, float v) {
    if (v >= 0.0f) atomicMax((int*)addr, __float_as_int(v));
    else           atomicMin((unsigned int*)addr, __float_as_uint(v));
}

// ---------------------------------------------------------------------------
// Degree helpers
// ---------------------------------------------------------------------------
__global__ void fill_ones_kernel(float* __restrict__ deg, int n) {
    int i = blockIdx.x * blockDim.x + threadIdx.x;
    if (i < n) deg[i] = 1.0f;                       // self-loop contribution
}

__global__ void count_deg_kernel(const int* __restrict__ col,
                                 float* __restrict__ deg, int e) {
    int i = blockIdx.x * blockDim.x + threadIdx.x;
    if (i < e) atomicAdd(&deg[col[i]], 1.0f);
}

__global__ void rsqrt_kernel(float* __restrict__ d, int n) {
    int i = blockIdx.x * blockDim.x + threadIdx.x;
    if (i < n) d[i] = rsqrtf(d[i]);                 // deg >= 1 always
}

// ---------------------------------------------------------------------------
// Fp32 GEMM via V_WMMA_F32_16X16X4_F32.
// C[M=32/block, 64] = A[M, K] * W[K, 64].  256 threads = 8 waves:
// wave w -> row-tile (w>>2), col-tile (w&3).  M % 32 == 0, K % 4 == 0.
// ---------------------------------------------------------------------------
template <int K>
__global__ void gemm_wmma_kernel(const float* __restrict__ A,
                                 const float* __restrict__ W,
                                 float* __restrict__ C) {
    const int NOUT = HID;                            // 64
    int wave  = threadIdx.x >> 5;
    int lane  = threadIdx.x & 31;
    int half  = lane >> 4;                           // 0: lanes 0-15, 1: 16-31
    int l     = lane & 15;
    int mbase = blockIdx.x * 32 + (wave >> 2) * 16;
    int n0    = (wave & 3) * 16;

    const float* arow = A + (size_t)(mbase + l) * K; // lane's A row
    v8f acc = {};
#pragma unroll
    for (int k0 = 0; k0 < K; k0 += 4) {
        int kk = k0 + half * 2;
        // A 16x4: lane holds A[row][kk], A[row][kk+1] (contiguous -> b64 load)
        v2f a = *reinterpret_cast<const v2f*>(arow + kk);
        // B 4x16: lane holds W[kk][n0+l], W[kk+1][n0+l]
        v2f b;
        b.x = W[kk * NOUT + n0 + l];
        b.y = W[(kk + 1) * NOUT + n0 + l];
        acc = __builtin_amdgcn_wmma_f32_16x16x4_f32(
            /*neg_a=*/false, a, /*neg_b=*/false, b,
            /*c_mod=*/(short)0, acc, /*reuse_a=*/false, /*reuse_b=*/false);
    }
    // C/D layout: VGPR r -> row (r + half*8), col (n0 + l)
    int crow = mbase + half * 8;
#pragma unroll
    for (int r = 0; r < 8; ++r)
        C[(size_t)(crow + r) * NOUT + n0 + l] = acc[r];
}

// ---------------------------------------------------------------------------
// agg[i] = dinv[i]^2 * h[i]   (self-loop message initializes the max)
// 16 threads per node, float4 per thread.
// ---------------------------------------------------------------------------
__global__ void selfloop_init_kernel(const float* __restrict__ h,
                                     const float* __restrict__ dinv,
                                     float* __restrict__ agg) {
    int t    = blockIdx.x * blockDim.x + threadIdx.x;   // N_NODES*16 threads
    int node = t >> 4;
    int f4   = (t & 15) << 2;
    float d  = dinv[node];
    float s  = d * d;
    float4 v = *reinterpret_cast<const float4*>(h + (size_t)node * HID + f4);
    float4 o;
    o.x = s * v.x; o.y = s * v.y; o.z = s * v.z; o.w = s * v.w;
    *reinterpret_cast<float4*>(agg + (size_t)node * HID + f4) = o;
}

// ---------------------------------------------------------------------------
// Edge scatter-max: 16 threads per edge, float4 gather + 4 atomic maxes.
// Gather of h[row] is fully coalesced (256B per edge); agg is L2-resident.
// ---------------------------------------------------------------------------
__global__ void scatter_max_kernel(const float* __restrict__ h,
                                   const int* __restrict__ row,
                                   const int* __restrict__ col,
                                   const float* __restrict__ dinv,
                                   float* __restrict__ agg) {
    int t  = blockIdx.x * blockDim.x + threadIdx.x;     // N_EDGES*16 threads
    int e  = t >> 4;
    int f4 = (t & 15) << 2;
    int r  = row[e];
    int c  = col[e];
    float norm = dinv[r] * dinv[c];
    float4 v = *reinterpret_cast<const float4*>(h + (size_t)r * HID + f4);
    float* dst = agg + (size_t)c * HID + f4;
    atomicMaxF(dst + 0, norm * v.x);
    atomicMaxF(dst + 1, norm * v.y);
    atomicMaxF(dst + 2, norm * v.z);
    atomicMaxF(dst + 3, norm * v.w);
}

// ---------------------------------------------------------------------------
// agg = relu(agg + b)   (in place, feature-major bias)
// ---------------------------------------------------------------------------
__global__ void bias_relu_kernel(float* __restrict__ agg,
                                 const float* __restrict__ b) {
    int i = blockIdx.x * blockDim.x + threadIdx.x;      // N_NODES*HID threads
    agg[i] = fmaxf(agg[i] + b[i & (HID - 1)], 0.0f);
}

// ---------------------------------------------------------------------------
// logits = h @ Wl + bl ; out = log_softmax(logits). One node per thread,
// Wl/bl staged in LDS (inner-loop reads are wave-broadcast).
// ---------------------------------------------------------------------------
__global__ void final_kernel(const float* __restrict__ h,
                             const float* __restrict__ Wl,
                             const float* __restrict__ bl,
                             float* __restrict__ out, int n) {
    __shared__ float sW[HID * N_CLASSES];
    __shared__ float sb[N_CLASSES];
    for (int i = threadIdx.x; i < HID * N_CLASSES; i += blockDim.x) sW[i] = Wl[i];
    if (threadIdx.x < N_CLASSES) sb[threadIdx.x] = bl[threadIdx.x];
    __syncthreads();

    int node = blockIdx.x * blockDim.x + threadIdx.x;
    if (node >= n) return;

    float logit[N_CLASSES];
#pragma unroll
    for (int c = 0; c < N_CLASSES; ++c) logit[c] = sb[c];

    const float* hr = h + (size_t)node * HID;
    for (int k = 0; k < HID; ++k) {
        float hv = hr[k];
#pragma unroll
        for (int c = 0; c < N_CLASSES; ++c)
            logit[c] = fmaf(hv, sW[k * N_CLASSES + c], logit[c]);
    }

    float m = logit[0];
#pragma unroll
    for (int c = 1; c < N_CLASSES; ++c) m = fmaxf(m, logit[c]);
    float s = 0.0f;
#pragma unroll
    for (int c = 0; c < N_CLASSES; ++c) s += __expf(logit[c] - m);
    float ls = __logf(s);

    float* orow = out + (size_t)node * N_CLASSES;
#pragma unroll
    for (int c = 0; c < N_CLASSES; ++c) orow[c] = logit[c] - m - ls;
}

// ---------------------------------------------------------------------------
extern "C" void kernel_launch(void* const* d_in, const int* in_sizes, int n_in,
                              void* d_out, int out_size, void* d_ws, size_t ws_size,
                              hipStream_t stream) {
    const float* x  = (const float*)d_in[0];
    const int*   ei = (const int*)d_in[1];
    const float* W1 = (const float*)d_in[2];
    const float* b1 = (const float*)d_in[3];
    const float* W2 = (const float*)d_in[4];
    const float* b2 = (const float*)d_in[5];
    const float* W3 = (const float*)d_in[6];
    const float* b3 = (const float*)d_in[7];
    const float* Wl = (const float*)d_in[8];
    const float* bl = (const float*)d_in[9];

    const int* row = ei;
    const int* col = ei + N_EDGES;

    // workspace: h [N*64] | agg [N*64] | dinv [N]   (~51.6 MB)
    float* h    = (float*)d_ws;
    float* agg  = h   + (size_t)N_NODES * HID;
    float* dinv = agg + (size_t)N_NODES * HID;
    float* outp = (float*)d_out;

    const int B = 256;

    // ---- degrees (shared across layers) ----
    fill_ones_kernel<<<(N_NODES + B - 1) / B, B, 0, stream>>>(dinv, N_NODES);
    count_deg_kernel<<<(N_EDGES + B - 1) / B, B, 0, stream>>>(col, dinv, N_EDGES);
    rsqrt_kernel<<<(N_NODES + B - 1) / B, B, 0, stream>>>(dinv, N_NODES);

    const int gemm_blocks    = N_NODES / 32;           // 3125 (exact)
    const int self_blocks    = (N_NODES * 16) / B;     // 6250 (exact)
    const int scatter_blocks = (N_EDGES * 16) / B;     // 100000 (exact)
    const int br_blocks      = (N_NODES * HID) / B;    // 25000 (exact)

    // ---- layer 1 ----
    gemm_wmma_kernel<IN_F><<<gemm_blocks, B, 0, stream>>>(x, W1, h);
    selfloop_init_kernel<<<self_blocks, B, 0, stream>>>(h, dinv, agg);
    scatter_max_kernel<<<scatter_blocks, B, 0, stream>>>(h, row, col, dinv, agg);
    bias_relu_kernel<<<br_blocks, B, 0, stream>>>(agg, b1);

    // ---- layer 2 ----
    gemm_wmma_kernel<HID><<<gemm_blocks, B, 0, stream>>>(agg, W2, h);
    selfloop_init_kernel<<<self_blocks, B, 0, stream>>>(h, dinv, agg);
    scatter_max_kernel<<<scatter_blocks, B, 0, stream>>>(h, row, col, dinv, agg);
    bias_relu_kernel<<<br_blocks, B, 0, stream>>>(agg, b2);

    // ---- layer 3 ----
    gemm_wmma_kernel<HID><<<gemm_blocks, B, 0, stream>>>(agg, W3, h);
    selfloop_init_kernel<<<self_blocks, B, 0, stream>>>(h, dinv, agg);
    scatter_max_kernel<<<scatter_blocks, B, 0, stream>>>(h, row, col, dinv, agg);
    bias_relu_kernel<<<br_blocks, B, 0, stream>>>(agg, b3);

    // ---- classifier + log_softmax ----
    final_kernel<<<(N_NODES + B - 1) / B, B, 0, stream>>>(agg, Wl, bl, outp, N_NODES);
}